// SemanticHypergraphModel_83966610636808
// MI455X (gfx1250) — compile-verified
//
#include <hip/hip_runtime.h>
#include <hip/hip_bf16.h>

// Problem constants (from reference setup_inputs)
#define BATCH 32
#define SEQ_L 4096
#define DIM_D 768
#define NTOPIC 2048
#define TOPK 16
#define NWORDS 24            // DIM_D / 32 bitmask words per topic

typedef float    v4f __attribute__((ext_vector_type(4)));
typedef unsigned v4u __attribute__((ext_vector_type(4)));
typedef int      v4i __attribute__((vector_size(16)));  // matches builtin param

#if defined(__gfx1250__) && \
    __has_builtin(__builtin_amdgcn_global_store_async_from_lds_b128) && \
    __has_builtin(__builtin_amdgcn_s_wait_asynccnt)
#define USE_ASYNC_LDS_STORE 1
#else
#define USE_ASYNC_LDS_STORE 0
#endif

// ---------------------------------------------------------------------------
// Kernel 1: per-topic top-16 along D, emitted as a membership bitmask.
// Layout: mask[w * NTOPIC + t] holds bits for rows d in [32w, 32w+32) of
// topic t (bit position == lane == d & 31).
// One wave32 per topic; 8 topics per 256-thread block.
// ---------------------------------------------------------------------------
__global__ void topk_mask_kernel(const float* __restrict__ tv,
                                 unsigned* __restrict__ mask) {
    const int wave = threadIdx.x >> 5;
    const int lane = threadIdx.x & 31;
    const int t = blockIdx.x * 8 + wave;

    // Coalesced load: lane holds d = j*32 + lane, j = 0..23
    float vals[NWORDS];
#pragma unroll
    for (int j = 0; j < NWORDS; ++j)
        vals[j] = tv[t * DIM_D + j * 32 + lane];

    unsigned chosen = 0;  // bit j: this lane's element (j*32+lane) selected
    for (int k = 0; k < TOPK; ++k) {
        float best = -__builtin_inff();
        int bidx = 0x7fffffff;
#pragma unroll
        for (int j = 0; j < NWORDS; ++j) {
            if (!((chosen >> j) & 1u) && vals[j] > best) {
                best = vals[j];
                bidx = j * 32 + lane;
            }
        }
        // wave32 argmax reduction; tie -> lower global index (matches top_k)
#pragma unroll
        for (int off = 16; off > 0; off >>= 1) {
            float ov = __shfl_xor(best, off, 32);
            int   oi = __shfl_xor(bidx, off, 32);
            if (ov > best || (ov == best && oi < bidx)) {
                best = ov;
                bidx = oi;
            }
        }
        if ((bidx & 31) == lane)
            chosen |= 1u << (bidx >> 5);
    }

    // word w of topic t is exactly a ballot of chosen-bit j across lanes
#pragma unroll
    for (int j = 0; j < NWORDS; ++j) {
        unsigned word = __builtin_amdgcn_ballot_w32(((chosen >> j) & 1u) != 0u);
        if (lane == 0)
            mask[j * NTOPIC + t] = word;
    }
}

// ---------------------------------------------------------------------------
// Kernel 2: materialize hypergraph [B, L, T]. One block per row l.
// Build the T-float row once in LDS, then broadcast to all 32 batches.
// Broadcast uses CDNA5 async LDS->global b128 stores (ASYNCcnt) when the
// builtin exists; otherwise nontemporal b128 stores from registers.
// ---------------------------------------------------------------------------
__global__ void fill_kernel(const unsigned* __restrict__ mask,
                            float* __restrict__ out) {
    __shared__ v4f row[NTOPIC / 4];  // 8 KB: one output row [T]

    const int l = blockIdx.x;
    const int tid = threadIdx.x;

    if (l < DIM_D) {
        const int w = l >> 5;
        const int bit = l & 31;
#pragma unroll
        for (int c = 0; c < 2; ++c) {
            const int q = tid + c * 256;  // quad index 0..511
            const v4u m = *(const v4u*)(mask + w * NTOPIC + q * 4);
            v4f v;
            v.x = (float)((m.x >> bit) & 1u);
            v.y = (float)((m.y >> bit) & 1u);
            v.z = (float)((m.z >> bit) & 1u);
            v.w = (float)((m.w >> bit) & 1u);
            row[q] = v;
        }
    } else {
#pragma unroll
        for (int c = 0; c < 2; ++c) {
            const int q = tid + c * 256;
            v4f z = {0.0f, 0.0f, 0.0f, 0.0f};
            row[q] = z;
        }
    }
    __syncthreads();

    const size_t rowbase = (size_t)l * NTOPIC;
    for (int b = 0; b < BATCH; ++b) {
        float* dst = out + (size_t)b * SEQ_L * NTOPIC + rowbase;
#pragma unroll
        for (int c = 0; c < 2; ++c) {
            const int q = tid + c * 256;
#if USE_ASYNC_LDS_STORE
            v4i* gp = (v4i*)(dst + q * 4);     // reinterpret pointee (generic)
            v4i* lp = (v4i*)&row[q];           // reinterpret pointee (generic)
            __builtin_amdgcn_global_store_async_from_lds_b128(
                (__attribute__((address_space(1))) v4i*)gp,
                (__attribute__((address_space(3))) v4i*)lp,
                /*offset=*/0, /*cpol=*/0);
#else
            __builtin_nontemporal_store(row[q], (v4f*)(dst + q * 4));
#endif
        }
    }
#if USE_ASYNC_LDS_STORE
    __builtin_amdgcn_s_wait_asynccnt(0);
#endif
}

// ---------------------------------------------------------------------------
extern "C" void kernel_launch(void* const* d_in, const int* in_sizes, int n_in,
                              void* d_out, int out_size, void* d_ws, size_t ws_size,
                              hipStream_t stream) {
    (void)in_sizes; (void)n_in; (void)out_size; (void)ws_size;
    // d_in[0] = x [B,L,D] (only shapes matter; values unused by the output)
    const float* topic_vectors = (const float*)d_in[1];  // [T, D] fp32
    float* out = (float*)d_out;                          // [B, L, T] fp32
    unsigned* mask = (unsigned*)d_ws;                    // [NWORDS, T] = 196.6 KB

    // 1) top-16 membership bitmask per topic (one wave32 per topic)
    topk_mask_kernel<<<NTOPIC / 8, 256, 0, stream>>>(topic_vectors, mask);

    // 2) stream the 1 GiB hypergraph: row computed once, broadcast x32
    fill_kernel<<<SEQ_L, 256, 0, stream>>>(mask, out);
}